// BottleneckSummary_12189117186367
// MI455X (gfx1250) — compile-verified
//
#include <hip/hip_runtime.h>
#include <hip/hip_bf16.h>
#include <math.h>

// Problem constants (match reference)
#define BS 16
#define TS 2048
#define HD 1024
#define MT_LEN 1024
#define TCH 8                 // t-chunks for deterministic partial reductions (TS/256)
#define SCALE_F (1.0f / 32.0f) // 1/sqrt(1024)
#define NEG_F (-10000.0f)

typedef __attribute__((ext_vector_type(2))) float v2f;
typedef __attribute__((ext_vector_type(8))) float v8f;

// ---------------------------------------------------------------------------
// Pass 1a: partial masked sums of hidden_states over t-chunks (deterministic)
// grid (HD/256, TCH, BS), block 256
__global__ __launch_bounds__(256)
void k_hs_partial(const float* __restrict__ hs, const int* __restrict__ att,
                  float* __restrict__ part) {
  const int h  = blockIdx.x * 256 + threadIdx.x;
  const int tc = blockIdx.y;
  const int b  = blockIdx.z;
  const float* base = hs + ((size_t)(b * TS + tc * 256)) * HD + h;
  const int*   am   = att + b * TS + tc * 256;
  float acc = 0.0f;
#pragma unroll 8
  for (int t = 0; t < 256; ++t)
    acc += base[(size_t)t * HD] * (float)am[t];
  part[(size_t)(b * TCH + tc) * HD + h] = acc;
}

// Pass 1b: attention-mask counts per batch. grid BS, block 256
__global__ __launch_bounds__(256)
void k_cnt(const int* __restrict__ att, float* __restrict__ cnt) {
  __shared__ float red[256];
  const int b = blockIdx.x, tid = threadIdx.x;
  float a = 0.0f;
#pragma unroll
  for (int i = 0; i < TS / 256; ++i) a += (float)att[b * TS + tid + i * 256];
  red[tid] = a; __syncthreads();
  for (int s = 128; s > 0; s >>= 1) { if (tid < s) red[tid] += red[tid + s]; __syncthreads(); }
  if (tid == 0) cnt[b] = red[0];
}

// Pass 1c: fixed-order reduction of TCH partials; optional divide by cnt[b].
// grid (HD/256, BS), block 256
__global__ __launch_bounds__(256)
void k_reduce8(const float* __restrict__ part, const float* __restrict__ cnt,
               float* __restrict__ out) {
  const int h = blockIdx.x * 256 + threadIdx.x;
  const int b = blockIdx.y;
  float acc = 0.0f;
#pragma unroll
  for (int tc = 0; tc < TCH; ++tc) acc += part[(size_t)(b * TCH + tc) * HD + h];
  if (cnt) acc /= cnt[b];
  out[(size_t)b * HD + h] = acc;
}

// ---------------------------------------------------------------------------
// WMMA GEMM: D(16xHD) = A(16xHD) x B(HDxHD) [+ bias], fp32 via V_WMMA_F32_16X16X4_F32.
// BT=true : B[k][n] = W[n*HD + k]   (i.e. multiply by W^T, W row-major)
// BT=false: B[k][n] = W[k*HD + n]   (multiply by W directly)
// One wave per 16x16 output tile. ISA 7.12.2 f32 layouts:
//   A 16x4: lane l(<16)->M=l, half=lane>>4 selects K pair {k0,k0+1}/{k0+2,k0+3}
//   B 4x16 (mirror): lane l -> N=l, half selects K pair
//   C/D:   VGPR r -> M = r + 8*half, lane l -> N = l
template <bool BT>
__global__ __launch_bounds__(32)
void k_gemm16(const float* __restrict__ A, const float* __restrict__ W,
              const float* __restrict__ bias, float* __restrict__ D) {
  const int lane = threadIdx.x;
  const int l    = lane & 15;
  const int half = lane >> 4;
  const int n0   = blockIdx.x * 16;
  const int n    = n0 + l;
  v8f c = {};
#pragma unroll 4
  for (int k0 = 0; k0 < HD; k0 += 4) {
    const int kb = k0 + half * 2;
    v2f a = *(const v2f*)(A + (size_t)l * HD + kb);
    v2f bm;
    if (BT) {
      bm = *(const v2f*)(W + (size_t)n * HD + kb);
    } else {
      bm.x = W[(size_t)kb * HD + n];
      bm.y = W[(size_t)(kb + 1) * HD + n];
    }
    c = __builtin_amdgcn_wmma_f32_16x16x4_f32(false, a, false, bm,
                                              (short)0, c, false, false);
  }
  const float bs = bias ? bias[n] : 0.0f;
#pragma unroll
  for (int r = 0; r < 8; ++r) {
    const int m = r + half * 8;
    D[(size_t)m * HD + n] = c[r] + bs;
  }
}

// ---------------------------------------------------------------------------
// qkbias[b] = dot(q[b], bk). grid BS, block 256
__global__ __launch_bounds__(256)
void k_qkbias(const float* __restrict__ q, const float* __restrict__ bk,
              float* __restrict__ qkb) {
  __shared__ float red[256];
  const int b = blockIdx.x, tid = threadIdx.x;
  float a = 0.0f;
#pragma unroll
  for (int i = 0; i < HD / 256; ++i) {
    const int h = tid + i * 256;
    a += q[(size_t)b * HD + h] * bk[h];
  }
  red[tid] = a; __syncthreads();
  for (int s = 128; s > 0; s >>= 1) { if (tid < s) red[tid] += red[tid + s]; __syncthreads(); }
  if (tid == 0) qkb[b] = red[0];
}

// ---------------------------------------------------------------------------
// scores[b,t] = att ? SCALE*(emb[b,t,:]·qk[b,:] + qkbias[b]) : NEG
// One wave per t row; qk[b] staged in LDS. grid (TS/8, BS), block 256
__global__ __launch_bounds__(256)
void k_scores(const float* __restrict__ emb, const int* __restrict__ att,
              const float* __restrict__ qk, const float* __restrict__ qkb,
              float* __restrict__ scores) {
  __shared__ float sqk[HD];
  const int b = blockIdx.y, tid = threadIdx.x;
  for (int i = tid; i < HD; i += 256) sqk[i] = qk[(size_t)b * HD + i];
  __syncthreads();
  const int wave = tid >> 5, lane = tid & 31;
  const int t = blockIdx.x * 8 + wave;
  const float* row = emb + ((size_t)b * TS + t) * HD;
  float acc = 0.0f;
#pragma unroll
  for (int i = 0; i < 8; ++i) {
    const int idx = i * 128 + lane * 4;
    float4 e = *(const float4*)(row + idx);
    acc += e.x * sqk[idx] + e.y * sqk[idx + 1] + e.z * sqk[idx + 2] + e.w * sqk[idx + 3];
  }
  for (int off = 16; off >= 1; off >>= 1) acc += __shfl_xor(acc, off, 32);
  if (lane == 0) {
    const float s = (att[b * TS + t] != 0) ? (acc + qkb[b]) * SCALE_F : NEG_F;
    scores[b * TS + t] = s;
  }
}

// ---------------------------------------------------------------------------
// Softmax over t per batch + write probas (ws) + mt/src proba outputs.
// grid BS, block 256 (8 elements/thread)
__global__ __launch_bounds__(256)
void k_softmax(const float* __restrict__ scores, const int* __restrict__ att,
               const int* __restrict__ fsm, float* __restrict__ probas,
               float* __restrict__ out_mt, float* __restrict__ out_src) {
  __shared__ float red[256];
  const int b = blockIdx.x, tid = threadIdx.x;
  float loc[TS / 256];
  float mx = -INFINITY;
#pragma unroll
  for (int i = 0; i < TS / 256; ++i) {
    loc[i] = scores[b * TS + tid + i * 256];
    mx = fmaxf(mx, loc[i]);
  }
  red[tid] = mx; __syncthreads();
  for (int s = 128; s > 0; s >>= 1) { if (tid < s) red[tid] = fmaxf(red[tid], red[tid + s]); __syncthreads(); }
  mx = red[0]; __syncthreads();
  float sum = 0.0f;
#pragma unroll
  for (int i = 0; i < TS / 256; ++i) { loc[i] = __expf(loc[i] - mx); sum += loc[i]; }
  red[tid] = sum; __syncthreads();
  for (int s = 128; s > 0; s >>= 1) { if (tid < s) red[tid] += red[tid + s]; __syncthreads(); }
  const float inv = 1.0f / red[0];
#pragma unroll
  for (int i = 0; i < TS / 256; ++i) {
    const int t = tid + i * 256;
    const float p = loc[i] * inv;
    probas[b * TS + t] = p;
    if (t < MT_LEN) {
      const float m = (float)(att[b * TS + t] * fsm[b * TS + t]);
      out_mt[b * MT_LEN + t] = p * m;
    } else {
      const float m = (float)(att[b * TS + t] * (1 - fsm[b * TS + t]));
      out_src[b * MT_LEN + (t - MT_LEN)] = p * m;
    }
  }
}

// ---------------------------------------------------------------------------
// Pass 2: partial weighted sums pe_part = sum_t probas[b,t]*emb[b,t,h] per t-chunk
// grid (HD/256, TCH, BS), block 256
__global__ __launch_bounds__(256)
void k_pe_partial(const float* __restrict__ emb, const float* __restrict__ probas,
                  float* __restrict__ part) {
  const int h  = blockIdx.x * 256 + threadIdx.x;
  const int tc = blockIdx.y;
  const int b  = blockIdx.z;
  const float* base = emb + ((size_t)(b * TS + tc * 256)) * HD + h;
  const float* pw   = probas + b * TS + tc * 256;
  float acc = 0.0f;
#pragma unroll 8
  for (int t = 0; t < 256; ++t)
    acc += base[(size_t)t * HD] * pw[t];
  part[(size_t)(b * TCH + tc) * HD + h] = acc;
}

// ---------------------------------------------------------------------------
extern "C" void kernel_launch(void* const* d_in, const int* in_sizes, int n_in,
                              void* d_out, int out_size, void* d_ws, size_t ws_size,
                              hipStream_t stream) {
  const float* hs  = (const float*)d_in[0];
  const float* emb = (const float*)d_in[1];
  const int*   att = (const int*)d_in[2];
  const int*   fsm = (const int*)d_in[3];
  // d_in[4] = first_piece_mask (unused by reference)
  const float* Wq = (const float*)d_in[5];
  const float* bq = (const float*)d_in[6];
  const float* Wk = (const float*)d_in[7];
  const float* bk = (const float*)d_in[8];
  const float* Wv = (const float*)d_in[9];
  const float* bv = (const float*)d_in[10];

  float* out_combined = (float*)d_out;                 // (16,1024)
  float* out_mt  = out_combined + BS * HD;             // (16,1024)
  float* out_src = out_mt + BS * MT_LEN;               // (16,1024)

  // Workspace layout (floats); total ~1.54 MB
  float* ws = (float*)d_ws;
  float* part_hs = ws;                          size_t o = (size_t)BS * TCH * HD;  // 131072
  float* avg     = ws + o;                      o += (size_t)BS * HD;              // +16384
  float* cnt     = ws + o;                      o += 16;
  float* q       = ws + o;                      o += (size_t)BS * HD;
  float* qk      = ws + o;                      o += (size_t)BS * HD;
  float* qkb     = ws + o;                      o += 16;
  float* scores  = ws + o;                      o += (size_t)BS * TS;
  float* probas  = ws + o;                      o += (size_t)BS * TS;
  float* part_pe = ws + o;                      o += (size_t)BS * TCH * HD;
  float* pe      = ws + o;                      o += (size_t)BS * HD;
  (void)ws_size; (void)n_in; (void)in_sizes; (void)out_size;

  // 1) masked average of hidden_states (deterministic partial-sum tree)
  k_hs_partial<<<dim3(HD / 256, TCH, BS), 256, 0, stream>>>(hs, att, part_hs);
  k_cnt<<<BS, 256, 0, stream>>>(att, cnt);
  k_reduce8<<<dim3(HD / 256, BS), 256, 0, stream>>>(part_hs, cnt, avg);

  // 2) q = avg @ Wq^T + bq        (WMMA f32, B = Wq transposed view)
  k_gemm16<true><<<HD / 16, 32, 0, stream>>>(avg, Wq, bq, q);

  // 3) qk = q @ Wk (folds Wk into q; exact re-association of q·k)
  k_gemm16<false><<<HD / 16, 32, 0, stream>>>(q, Wk, nullptr, qk);
  k_qkbias<<<BS, 256, 0, stream>>>(q, bk, qkb);

  // 4) scores = SCALE*(emb·qk + q·bk), masked    (streaming pass #1 over emb)
  k_scores<<<dim3(TS / 8, BS), 256, 0, stream>>>(emb, att, qk, qkb, scores);

  // 5) softmax + masked proba outputs
  k_softmax<<<BS, 256, 0, stream>>>(scores, att, fsm, probas, out_mt, out_src);

  // 6) pe = sum_t p*emb                           (streaming pass #2 over emb)
  k_pe_partial<<<dim3(HD / 256, TCH, BS), 256, 0, stream>>>(emb, probas, part_pe);
  k_reduce8<<<dim3(HD / 256, BS), 256, 0, stream>>>(part_pe, nullptr, pe);

  // 7) combined = pe @ Wv^T + bv  (softmax probs sum to 1, so bv carries through)
  k_gemm16<true><<<HD / 16, 32, 0, stream>>>(pe, Wv, bv, out_combined);
}